// LinearCausalSCMLayer_64321430224944
// MI455X (gfx1250) — compile-verified
//
#include <hip/hip_runtime.h>
#include <math.h>

#if defined(__HIP_DEVICE_COMPILE__)
#if !__has_builtin(__builtin_amdgcn_wmma_f32_16x16x4_f32)
#error "gfx1250 wmma f32 16x16x4 builtin not available on this toolchain (device pass)"
#endif
#endif

typedef float v2f __attribute__((ext_vector_type(2)));
typedef float v8f __attribute__((ext_vector_type(8)));

constexpr int KDIM  = 1024;     // nodes (K)
constexpr int VDIM  = 2;        // intervention variants
constexpr int MT    = 64;       // samples (rows) per workgroup
constexpr int NT    = 128;      // threads per workgroup (4 wave32)
constexpr int US    = MT + 1;   // u_lds row stride (pad to dodge bank conflicts)
constexpr int CHUNK = 128;      // k-values of A staged per barrier phase
constexpr int ATS   = 16 + 1;   // at_lds stride

// u_lds  : [KDIM][US]  transposed state tile, pre-seeded with z, solved in place
// at_lds : [CHUNK][ATS] staged A^T chunk: at[k - cs][c] = A[jb + c][k]
__global__ __launch_bounds__(NT, 1)
void scm_forward_kernel(const float* __restrict__ z,
                        const int*   __restrict__ target,
                        const int*   __restrict__ variant,
                        const float* __restrict__ Araw,
                        const float* __restrict__ means,
                        const float* __restrict__ logsc,
                        float* __restrict__ out)
{
  extern __shared__ float smem[];
  float* u_lds  = smem;                 // KDIM * US floats (~260 KB)
  float* at_lds = smem + KDIM * US;     // CHUNK * ATS floats (~8.5 KB)

  const int tid  = threadIdx.x;
  const int lane = tid & 31;
  const int wv   = tid >> 5;            // wave id 0..3 -> 16-row strip
  const int m    = lane & 15;
  const int koff = (lane < 16) ? 0 : 2; // K split across half-waves (f32 WMMA layout)
  const int rb   = wv * 16;
  const int b0   = blockIdx.x * MT;

  // ---- seed u_lds with z (coalesced global read, transposed LDS write) ----
  for (int i = tid; i < MT * KDIM; i += NT) {
    int r = i >> 10;                    // i / KDIM
    int k = i & (KDIM - 1);
    u_lds[k * US + r] = z[(size_t)(b0 + r) * KDIM + k];
  }

  // ---- hoist per-sample intervention constants ----
  int   tr  = -1;
  float ivm = 0.f, ivs = 0.f, zt = 0.f;
  if (tid < MT) {
    int b = b0 + tid;
    tr = target[b];
    if (tr >= 0) {
      int vt = variant[b];
      ivm = means[tr * VDIM + vt];
      ivs = expf(logsc[tr * VDIM + vt]);
      zt  = z[(size_t)b * KDIM + tr];
    }
  }

  // ---- blocked forward substitution over K ----
  for (int t = 0; t < KDIM / 16; ++t) {
    const int jb      = t * 16;
    const int kmax    = jb + 16;                       // need A cols [0, jb+16)
    const int cs_last = ((kmax - 1) / CHUNK) * CHUNK;  // chunk holding diagonal block

    v8f acc = {};                                      // 16x16 f32 accumulator tile
    for (int cs = 0; cs < kmax; cs += CHUNK) {
      const int ce   = (cs + CHUNK < kmax) ? (cs + CHUNK) : kmax;
      const int clen = ce - cs;
      __syncthreads();                                 // prior consumers of at_lds/u_lds done
      // stage A[jb..jb+15][cs..ce) transposed (coalesced along k)
      for (int i = tid; i < 16 * clen; i += NT) {
        int c  = i / clen;
        int kk = i - c * clen;
        at_lds[kk * ATS + c] = Araw[(size_t)(jb + c) * KDIM + cs + kk];
      }
      // prefetch next block step's A rows (hides L2/HBM latency behind GEMM)
      if (tid < 16 && jb + 16 < KDIM)
        __builtin_prefetch(Araw + (size_t)(jb + 16 + tid) * KDIM + cs, 0, 1);
      __syncthreads();
      // GEMM part: acc += U[rb.., cs..min(ce,jb)) * A^T  via f32 WMMA 16x16x4
      // range length is always a multiple of 16 -> exactly 4 WMMA steps per iter
      const int ge = (jb < ce) ? jb : ce;
      for (int k16 = cs; k16 < ge; k16 += 16) {
        #pragma unroll
        for (int s = 0; s < 4; ++s) {
          const int k  = k16 + s * 4;
          const int ab = (k + koff) * US + rb + m;        // A pair base (stride US)
          const int bb = (k - cs + koff) * ATS + m;       // B pair base (stride ATS)
          v2f a, b;
          a.x = u_lds[ab];
          a.y = u_lds[ab + US];
          b.x = at_lds[bb];
          b.y = at_lds[bb + ATS];
          acc = __builtin_amdgcn_wmma_f32_16x16x4_f32(false, a, false, b,
                                                      (short)0, acc, false, false);
        }
      }
    }

    // ---- fold accumulator into u_lds (which already holds z): partial = z + GEMM ----
    {
      const int col = jb + m;
      const int row = rb + ((lane < 16) ? 0 : 8);      // C/D layout: vgpr v -> M = v (+8)
      #pragma unroll
      for (int v = 0; v < 8; ++v)
        u_lds[col * US + row + v] += acc[v];
    }
    __syncthreads();

    // ---- 16 serial in-block steps + intervention overwrite (one thread per row) ----
    if (tid < MT) {
      const int r = tid;
      float vals[16];
      #pragma unroll
      for (int c = 0; c < 16; ++c) {
        float v = u_lds[(jb + c) * US + r];
        #pragma unroll
        for (int i = 0; i < c; ++i)                    // A[jb+c][jb+i] from staged chunk
          v = fmaf(at_lds[(jb + i - cs_last) * ATS + c], vals[i], v);
        if (tr == jb + c) v = fmaf(ivs, zt, ivm);      // overwrite feeds downstream
        vals[c] = v;
        u_lds[(jb + c) * US + r] = v;
      }
    }
    // next iteration's first __syncthreads orders solve writes vs. GEMM reads
  }

  __syncthreads();
  // ---- write solved tile out (coalesced) ----
  for (int i = tid; i < MT * KDIM; i += NT) {
    int r = i >> 10;
    int k = i & (KDIM - 1);
    out[(size_t)(b0 + r) * KDIM + k] = u_lds[k * US + r];
  }
}

extern "C" void kernel_launch(void* const* d_in, const int* in_sizes, int n_in,
                              void* d_out, int out_size, void* d_ws, size_t ws_size,
                              hipStream_t stream) {
  const float* zp   = (const float*)d_in[0];
  const int*   tg   = (const int*)  d_in[1];
  const int*   vr   = (const int*)  d_in[2];
  const float* Ar   = (const float*)d_in[3];
  const float* mn   = (const float*)d_in[4];
  const float* ls   = (const float*)d_in[5];
  float*       outp = (float*)d_out;

  const int B = in_sizes[1];                       // 8192 samples
  const size_t shmem = (size_t)(KDIM * US + CHUNK * ATS) * sizeof(float); // ~275 KB

  (void)hipFuncSetAttribute((const void*)scm_forward_kernel,
                            hipFuncAttributeMaxDynamicSharedMemorySize, (int)shmem);

  scm_forward_kernel<<<dim3(B / MT), dim3(NT), shmem, stream>>>(
      zp, tg, vr, Ar, mn, ls, outp);
}